// SyntaxAttnDecoderRNN_1_4200478016030
// MI455X (gfx1250) — compile-verified
//
#include <hip/hip_runtime.h>
#include <hip/hip_bf16.h>
#include <math.h>

#define H 1024
#define V 50257
#define L 512

typedef __attribute__((ext_vector_type(2))) float v2f;
typedef __attribute__((ext_vector_type(4))) float v4f;
typedef __attribute__((ext_vector_type(8))) float v8f;

__device__ __forceinline__ float waveReduceSum(float v) {
#pragma unroll
  for (int off = 16; off > 0; off >>= 1) v += __shfl_xor(v, off, 32);
  return v;
}

// ---------------------------------------------------------------------------
// K1: t1 = tanh(pt_dense_W @ [embedded;h0])   (rows 0..H-1)
//     hWb = attn_W[:, :H] @ h0 + attn_b       (rows H..2H-1)
// one wave32 per output row
// ---------------------------------------------------------------------------
__global__ void k1_pt_and_hwb(const int* __restrict__ token,
                              const float* __restrict__ hidden,
                              const float* __restrict__ emb,
                              const float* __restrict__ ptW,
                              const float* __restrict__ attnW,
                              const float* __restrict__ attnB,
                              float* __restrict__ t1,
                              float* __restrict__ hWb) {
  int wave = threadIdx.x >> 5, lane = threadIdx.x & 31;
  int row = blockIdx.x * 8 + wave;            // 0..2047
  int tok = token[0];
  if (row < H) {
    const float* w = ptW + (size_t)row * (2 * H);
    const float* e = emb + (size_t)tok * H;
    float acc = 0.f;
    for (int i = lane; i < H; i += 32) acc += w[i] * e[i] + w[H + i] * hidden[i];
    acc = waveReduceSum(acc);
    if (lane == 0) t1[row] = tanhf(acc);
  } else {
    int r = row - H;
    const float* w = attnW + (size_t)r * (2 * H);
    float acc = 0.f;
    for (int i = lane; i < H; i += 32) acc += w[i] * hidden[i];
    acc = waveReduceSum(acc);
    if (lane == 0) hWb[r] = acc + attnB[r];
  }
}

// ---------------------------------------------------------------------------
// K2: p = sigmoid(pt_dot_W @ t1); p_t = clip(round(s_len*p)); syntax mask;
//     zero the score accumulator. 1 block, 512 threads.
// ---------------------------------------------------------------------------
__global__ void k2_pt_syntax(const float* __restrict__ ptDotW,
                             const float* __restrict__ t1,
                             const int* __restrict__ sLen,
                             const int* __restrict__ synMat,
                             float* __restrict__ syntax,
                             float* __restrict__ scores) {
  __shared__ float red[512];
  __shared__ int sp;
  int tid = threadIdx.x;
  float acc = 0.f;
  for (int i = tid; i < H; i += 512) acc += ptDotW[i] * t1[i];
  red[tid] = acc; __syncthreads();
  for (int s = 256; s > 0; s >>= 1) { if (tid < s) red[tid] += red[tid + s]; __syncthreads(); }
  if (tid == 0) {
    float p = 1.f / (1.f + expf(-red[0]));
    int pt = (int)rintf((float)sLen[0] * p);    // round-half-even, like jnp.round
    pt = pt < 0 ? 0 : (pt > L - 1 ? L - 1 : pt);
    sp = pt;
  }
  __syncthreads();
  float d = (float)synMat[sp * L + tid];
  syntax[tid] = (d > 4.0f) ? 0.f : expf(-(d * d) * 0.125f);  // 2*sigma^2 = 8
  scores[tid] = 0.f;
}

// ---------------------------------------------------------------------------
// K3: attention energy GEMM on the WMMA pipe (fp32 -> fp32, no downcast):
//   E = enc[512,1024] @ W2t[1024,1024]  with W2[k][n] = attn_W[n][H+k]
//   scores[m] += sum_n  relu(E[m][n] + hWb[n]) * v[n]
// One wave per 16x16 output tile; 256x v_wmma_f32_16x16x4_f32 along K.
// ---------------------------------------------------------------------------
__global__ void k3_attn_wmma(const float* __restrict__ enc,
                             const float* __restrict__ attnW,
                             const float* __restrict__ hWb,
                             const float* __restrict__ vvec,
                             float* __restrict__ scores) {
  int wave = threadIdx.x >> 5, lane = threadIdx.x & 31;
  int tile = blockIdx.x * 8 + wave;           // 0..2047  (32 x 64 tiles)
  int tm = tile >> 6, tn = tile & 63;
  int half = lane >> 4, nrow = lane & 15;

  // A fragment (16x4 f32): lane(h,m) holds A[m][2h], A[m][2h+1]
  const float* arow = enc + (size_t)(tm * 16 + nrow) * H + 2 * half;
  // B fragment (4x16 f32): lane(h,n) holds B[2h][n], B[2h+1][n]
  //   B[k][n] = attn_W[n][H+k]  -> consecutive along attn_W row
  const float* brow = attnW + (size_t)(tn * 16 + nrow) * (2 * H) + H + 2 * half;

  v8f acc = {};
  for (int k0 = 0; k0 < H; k0 += 4) {
    v2f a = *(const v2f*)(arow + k0);
    v2f b = *(const v2f*)(brow + k0);
    acc = __builtin_amdgcn_wmma_f32_16x16x4_f32(
        /*neg_a=*/false, a, /*neg_b=*/false, b,
        /*c_mod=*/(short)0, acc, /*reuse_a=*/false, /*reuse_b=*/false);
  }

  // Fused epilogue: C/D layout -> lane holds (M = r + 8*half, N = nrow)
  int N = tn * 16 + nrow;
  float hb = hWb[N], vv = vvec[N];
#pragma unroll
  for (int r = 0; r < 8; ++r) {
    float val = fmaxf(acc[r] + hb, 0.f) * vv;
#pragma unroll
    for (int off = 1; off < 16; off <<= 1) val += __shfl_xor(val, off, 32);
    if (nrow == 0) atomicAdd(&scores[tm * 16 + r + 8 * half], val);
  }
}

// ---------------------------------------------------------------------------
// K4: softmax(scores) * syntax, renormalize -> aw. 1 block, 512 threads.
// ---------------------------------------------------------------------------
__global__ void k4_softmax(const float* __restrict__ scores,
                           const float* __restrict__ syntax,
                           float* __restrict__ aw,
                           float* __restrict__ outAw) {
  __shared__ float red[512];
  int tid = threadIdx.x;
  float s = scores[tid];
  red[tid] = s; __syncthreads();
  for (int st = 256; st > 0; st >>= 1) { if (tid < st) red[tid] = fmaxf(red[tid], red[tid + st]); __syncthreads(); }
  float mx = red[0]; __syncthreads();
  float e = expf(s - mx);
  red[tid] = e; __syncthreads();
  for (int st = 256; st > 0; st >>= 1) { if (tid < st) red[tid] += red[tid + st]; __syncthreads(); }
  float sum = red[0]; __syncthreads();
  float a = (e / sum) * syntax[tid];
  red[tid] = a; __syncthreads();
  for (int st = 256; st > 0; st >>= 1) { if (tid < st) red[tid] += red[tid + st]; __syncthreads(); }
  float w = a / red[0];
  aw[tid] = w;
  outAw[tid] = w;
}

// K5: context[h] = sum_l aw[l]*enc[l][h]   (coalesced over h)
__global__ void k5_context(const float* __restrict__ aw,
                           const float* __restrict__ enc,
                           float* __restrict__ context) {
  int h = blockIdx.x * blockDim.x + threadIdx.x;
  float acc = 0.f;
  for (int l = 0; l < L; ++l) acc += aw[l] * enc[(size_t)l * H + h];
  context[h] = acc;
}

// K6: x = relu(comb_W @ [embedded;context] + comb_b); wave per row
__global__ void k6_comb(const int* __restrict__ token,
                        const float* __restrict__ emb,
                        const float* __restrict__ context,
                        const float* __restrict__ combW,
                        const float* __restrict__ combB,
                        float* __restrict__ x) {
  int wave = threadIdx.x >> 5, lane = threadIdx.x & 31;
  int row = blockIdx.x * 8 + wave;
  int tok = token[0];
  const float* w = combW + (size_t)row * (2 * H);
  const float* e = emb + (size_t)tok * H;
  float acc = 0.f;
  for (int i = lane; i < H; i += 32) acc += w[i] * e[i] + w[H + i] * context[i];
  acc = waveReduceSum(acc);
  if (lane == 0) x[row] = fmaxf(acc + combB[row], 0.f);
}

// K7: gi = W_ih@x + b_ih (rows 0..3071), gh = W_hh@h0 + b_hh (rows 3072..6143)
__global__ void k7_gru_mv(const float* __restrict__ Wih,
                          const float* __restrict__ Whh,
                          const float* __restrict__ bih,
                          const float* __restrict__ bhh,
                          const float* __restrict__ x,
                          const float* __restrict__ hidden,
                          float* __restrict__ gi, float* __restrict__ gh) {
  int wave = threadIdx.x >> 5, lane = threadIdx.x & 31;
  int row = blockIdx.x * 8 + wave;            // 0..6143
  float acc = 0.f;
  if (row < 3 * H) {
    const float* w = Wih + (size_t)row * H;
    for (int i = lane; i < H; i += 32) acc += w[i] * x[i];
    acc = waveReduceSum(acc);
    if (lane == 0) gi[row] = acc + bih[row];
  } else {
    int r = row - 3 * H;
    const float* w = Whh + (size_t)r * H;
    for (int i = lane; i < H; i += 32) acc += w[i] * hidden[i];
    acc = waveReduceSum(acc);
    if (lane == 0) gh[r] = acc + bhh[r];
  }
}

// K8: GRU gates -> h_new. 1 block, 1024 threads.
__global__ void k8_gates(const float* __restrict__ gi, const float* __restrict__ gh,
                         const float* __restrict__ hidden,
                         float* __restrict__ hnew, float* __restrict__ outH) {
  int h = threadIdx.x;
  float r = 1.f / (1.f + expf(-(gi[h] + gh[h])));
  float z = 1.f / (1.f + expf(-(gi[H + h] + gh[H + h])));
  float n = tanhf(gi[2 * H + h] + r * gh[2 * H + h]);
  float hv = (1.f - z) * n + z * hidden[h];
  hnew[h] = hv;
  outH[h] = hv;
}

// ---------------------------------------------------------------------------
// K9: logits = out_W @ h_new + out_b.  THE bandwidth stage (206 MB of out_W).
// One wave32 per vocab row; b128 loads -> each wave streams 512B/iter.
// ---------------------------------------------------------------------------
__global__ void k9_logits(const float* __restrict__ outW,
                          const float* __restrict__ outB,
                          const float* __restrict__ hnew,
                          float* __restrict__ logits) {
  int wave = threadIdx.x >> 5, lane = threadIdx.x & 31;
  int row = blockIdx.x * 8 + wave;
  if (row >= V) return;
  const float* w = outW + (size_t)row * H;
  float acc = 0.f;
#pragma unroll
  for (int c = 0; c < 8; ++c) {
    int idx = c * 128 + lane * 4;
    v4f wv = *(const v4f*)(w + idx);
    v4f hv = *(const v4f*)(hnew + idx);
    acc += wv.x * hv.x + wv.y * hv.y + wv.z * hv.z + wv.w * hv.w;
  }
  acc = waveReduceSum(acc);
  if (lane == 0) logits[row] = acc + outB[row];
}

// K10: reduce max and log-sum-exp over V. 1 block, 1024 threads.
__global__ void k10_lse(const float* __restrict__ logits, float* __restrict__ red2) {
  __shared__ float red[1024];
  int tid = threadIdx.x;
  float mx = -INFINITY;
  for (int i = tid; i < V; i += 1024) mx = fmaxf(mx, logits[i]);
  red[tid] = mx; __syncthreads();
  for (int s = 512; s > 0; s >>= 1) { if (tid < s) red[tid] = fmaxf(red[tid], red[tid + s]); __syncthreads(); }
  mx = red[0]; __syncthreads();
  float sum = 0.f;
  for (int i = tid; i < V; i += 1024) sum += expf(logits[i] - mx);
  red[tid] = sum; __syncthreads();
  for (int s = 512; s > 0; s >>= 1) { if (tid < s) red[tid] += red[tid + s]; __syncthreads(); }
  if (tid == 0) { red2[0] = mx; red2[1] = logf(red[0]); }
}

// K11: apply log_softmax in place on d_out logits region
__global__ void k11_apply(float* __restrict__ logits, const float* __restrict__ red2) {
  int i = blockIdx.x * blockDim.x + threadIdx.x;
  if (i < V) logits[i] = logits[i] - red2[0] - red2[1];
}

// ---------------------------------------------------------------------------
extern "C" void kernel_launch(void* const* d_in, const int* in_sizes, int n_in,
                              void* d_out, int out_size, void* d_ws, size_t ws_size,
                              hipStream_t stream) {
  const int*   token  = (const int*)  d_in[0];
  const int*   sLen   = (const int*)  d_in[1];
  const int*   synMat = (const int*)  d_in[2];
  const float* hidden = (const float*)d_in[3];
  const float* enc    = (const float*)d_in[4];
  const float* emb    = (const float*)d_in[5];
  const float* ptW    = (const float*)d_in[6];
  const float* ptDotW = (const float*)d_in[7];
  const float* attnW  = (const float*)d_in[8];
  const float* attnB  = (const float*)d_in[9];
  const float* vvec   = (const float*)d_in[10];
  const float* combW  = (const float*)d_in[11];
  const float* combB  = (const float*)d_in[12];
  const float* Wih    = (const float*)d_in[13];
  const float* Whh    = (const float*)d_in[14];
  const float* bih    = (const float*)d_in[15];
  const float* bhh    = (const float*)d_in[16];
  const float* outW   = (const float*)d_in[17];
  const float* outB   = (const float*)d_in[18];

  float* out = (float*)d_out;
  float* outLogits = out;              // [V]
  float* outH      = out + V;          // [H]
  float* outAw     = out + V + H;      // [L]

  float* ws = (float*)d_ws;
  float* t1      = ws;                 // [H]
  float* hWb     = ws + 1024;          // [H]
  float* syntax  = ws + 2048;          // [L]
  float* scores  = ws + 2560;          // [L]
  float* aw      = ws + 3072;          // [L]
  float* context = ws + 3584;          // [H]
  float* x       = ws + 4608;          // [H]
  float* gi      = ws + 5632;          // [3H]
  float* gh      = ws + 8704;          // [3H]
  float* hnew    = ws + 11776;         // [H]
  float* red2    = ws + 12800;         // [2]

  k1_pt_and_hwb<<<256, 256, 0, stream>>>(token, hidden, emb, ptW, attnW, attnB, t1, hWb);
  k2_pt_syntax<<<1, 512, 0, stream>>>(ptDotW, t1, sLen, synMat, syntax, scores);
  k3_attn_wmma<<<256, 256, 0, stream>>>(enc, attnW, hWb, vvec, scores);
  k4_softmax<<<1, 512, 0, stream>>>(scores, syntax, aw, outAw);
  k5_context<<<4, 256, 0, stream>>>(aw, enc, context);
  k6_comb<<<128, 256, 0, stream>>>(token, emb, context, combW, combB, x);
  k7_gru_mv<<<768, 256, 0, stream>>>(Wih, Whh, bih, bhh, x, hidden, gi, gh);
  k8_gates<<<1, 1024, 0, stream>>>(gi, gh, hidden, hnew, outH);
  k9_logits<<<(V + 7) / 8, 256, 0, stream>>>(outW, outB, hnew, outLogits);
  k10_lse<<<1, 1024, 0, stream>>>(outLogits, red2);
  k11_apply<<<(V + 255) / 256, 256, 0, stream>>>(outLogits, red2);
}